// TsStddev_17051020165339
// MI455X (gfx1250) — compile-verified
//
#include <hip/hip_runtime.h>
#include <stdint.h>

#define T_IN   4096
#define WIN    20
#define T_OUT  4077           // (4096 - 20) / 1 + 1
#define NROWS  (256 * 64)
#define BLOCK  256
#define CHUNK  16             // elements scanned per thread (256*16 = 4096)

// ---- CDNA5 async global->LDS copy (ASYNCcnt path), via inline asm ----------
// LDS destination address is taken from the VDST operand; the low 32 bits of a generic pointer to a
// __shared__ object are the LDS byte offset (aperture scheme, ISA ch.10.2).
__device__ __forceinline__ void async_load_b128_to_lds(unsigned lds_byte_off,
                                                       const void* gptr) {
  asm volatile("global_load_async_to_lds_b128 %0, %1, off"
               :
               : "v"(lds_byte_off), "v"((unsigned long long)(uintptr_t)gptr)
               : "memory");
}

__device__ __forceinline__ void wait_asynccnt0() {
  asm volatile("s_wait_asynccnt 0" ::: "memory");
}

__global__ __launch_bounds__(BLOCK) void ts_stddev_kernel(
    const float* __restrict__ x, float* __restrict__ out) {
  __shared__ __align__(16) float sx[T_IN];        // staged row
  __shared__ __align__(16) float sc1[T_IN + 4];   // exclusive cumsum of x   [0..4096]
  __shared__ __align__(16) float sc2[T_IN + 4];   // exclusive cumsum of x^2 [0..4096]
  __shared__ float wt1[BLOCK / 32];
  __shared__ float wt2[BLOCK / 32];

  const int t   = threadIdx.x;
  const int row = blockIdx.x;
  const float* xrow = x + (size_t)row * T_IN;

  // ---- Stage 1: async copy whole row (16 KB) global -> LDS -----------------
#pragma unroll
  for (int r = 0; r < T_IN / (BLOCK * 4); ++r) {  // 4 rounds x 256 lanes x 16B
    const int elem = (t + r * BLOCK) * 4;
    const unsigned lds_off = (unsigned)(uintptr_t)(&sx[elem]);
    async_load_b128_to_lds(lds_off, xrow + elem);
  }
  wait_asynccnt0();
  __syncthreads();

  // ---- Stage 2: exclusive prefix sums of x and x^2 in LDS ------------------
  const int base = t * CHUNK;
  float xv[CHUNK];
#pragma unroll
  for (int q = 0; q < CHUNK / 4; ++q) {           // ds_load_b128 x4
    const float4 v4 = reinterpret_cast<const float4*>(sx + base)[q];
    xv[4 * q + 0] = v4.x; xv[4 * q + 1] = v4.y;
    xv[4 * q + 2] = v4.z; xv[4 * q + 3] = v4.w;
  }
  float e1[CHUNK], e2[CHUNK];
  float s1 = 0.0f, s2 = 0.0f;
#pragma unroll
  for (int j = 0; j < CHUNK; ++j) {               // per-thread exclusive scan
    e1[j] = s1; e2[j] = s2;
    s1 += xv[j];
    s2 = fmaf(xv[j], xv[j], s2);
  }

  // wave32 inclusive scan of per-thread chunk totals
  float w1 = s1, w2 = s2;
  const int lane = t & 31;
#pragma unroll
  for (int d = 1; d < 32; d <<= 1) {
    const float u1 = __shfl_up(w1, (unsigned)d, 32);
    const float u2 = __shfl_up(w2, (unsigned)d, 32);
    if (lane >= d) { w1 += u1; w2 += u2; }
  }
  if (lane == 31) { wt1[t >> 5] = w1; wt2[t >> 5] = w2; }
  __syncthreads();

  float off1 = 0.0f, off2 = 0.0f;
#pragma unroll
  for (int w = 0; w < BLOCK / 32; ++w) {          // offsets from earlier waves
    const bool take = w < (t >> 5);
    off1 += take ? wt1[w] : 0.0f;
    off2 += take ? wt2[w] : 0.0f;
  }
  const float b1 = off1 + (w1 - s1);              // thread's exclusive base
  const float b2 = off2 + (w2 - s2);

#pragma unroll
  for (int q = 0; q < CHUNK / 4; ++q) {           // ds_store_b128 x8
    float4 o1, o2;
    o1.x = b1 + e1[4 * q + 0]; o1.y = b1 + e1[4 * q + 1];
    o1.z = b1 + e1[4 * q + 2]; o1.w = b1 + e1[4 * q + 3];
    o2.x = b2 + e2[4 * q + 0]; o2.y = b2 + e2[4 * q + 1];
    o2.z = b2 + e2[4 * q + 2]; o2.w = b2 + e2[4 * q + 3];
    reinterpret_cast<float4*>(sc1 + base)[q] = o1;
    reinterpret_cast<float4*>(sc2 + base)[q] = o2;
  }
  if (t == BLOCK - 1) {                           // grand totals at index 4096
    sc1[T_IN] = b1 + s1;      sc2[T_IN] = b2 + s2;
    sc1[T_IN + 1] = 0.0f;     sc2[T_IN + 1] = 0.0f;
    sc1[T_IN + 2] = 0.0f;     sc2[T_IN + 2] = 0.0f;
    sc1[T_IN + 3] = 0.0f;     sc2[T_IN + 3] = 0.0f;
  }
  __syncthreads();

  // ---- Stage 3: windowed stddev, 4 outputs/thread/iter, NT stores ----------
  float* orow = out + (size_t)row * T_OUT;
  const float inv_n   = 1.0f / (float)WIN;
  const float inv_nm1 = 1.0f / (float)(WIN - 1);
#pragma unroll
  for (int i = 0; i < 4; ++i) {
    const int p0 = 4 * (t + BLOCK * i);           // 16B aligned; p0+20 also
    if (p0 < T_OUT) {
      const float4 lo1 = *reinterpret_cast<const float4*>(sc1 + p0);
      const float4 hi1 = *reinterpret_cast<const float4*>(sc1 + p0 + WIN);
      const float4 lo2 = *reinterpret_cast<const float4*>(sc2 + p0);
      const float4 hi2 = *reinterpret_cast<const float4*>(sc2 + p0 + WIN);
      const float S1[4] = {hi1.x - lo1.x, hi1.y - lo1.y,
                           hi1.z - lo1.z, hi1.w - lo1.w};
      const float S2[4] = {hi2.x - lo2.x, hi2.y - lo2.y,
                           hi2.z - lo2.z, hi2.w - lo2.w};
#pragma unroll
      for (int j = 0; j < 4; ++j) {
        const int p = p0 + j;
        if (p < T_OUT) {
          const float var = (S2[j] - S1[j] * S1[j] * inv_n) * inv_nm1;
          const float sd  = sqrtf(fmaxf(var, 0.0f));
          __builtin_nontemporal_store(sd, orow + p);   // streamed once, keep L2 clean
        }
      }
    }
  }
}

extern "C" void kernel_launch(void* const* d_in, const int* in_sizes, int n_in,
                              void* d_out, int out_size, void* d_ws, size_t ws_size,
                              hipStream_t stream) {
  (void)in_sizes; (void)n_in; (void)d_ws; (void)ws_size; (void)out_size;
  const float* x = (const float*)d_in[0];
  float* out = (float*)d_out;
  ts_stddev_kernel<<<dim3(NROWS), dim3(BLOCK), 0, stream>>>(x, out);
}